// ChebLayer_65919158059656
// MI455X (gfx1250) — compile-verified
//
#include <hip/hip_runtime.h>
#include <hip/hip_bf16.h>

#define N_NODES 50000
#define N_EDGES 800000
#define D 128
#define KTOT 384            // concatenated K: [feature | ax | ax2]
#define BN_EPS 1e-5f
#define NTILES (N_NODES / 16)   // 3125 exactly

typedef float v2f __attribute__((ext_vector_type(2)));
typedef float v8f __attribute__((ext_vector_type(8)));

// ---- workspace layout (float offsets, all 16B-aligned) ----
#define OFF_DEG    0
#define OFF_NORM   (N_NODES)                  //  50,000
#define OFF_AGG1   (2 * N_NODES)              // 100,000
#define OFF_AGG2   (OFF_AGG1 + N_NODES * D)   // 6,500,000
#define OFF_WPK    (OFF_AGG2 + N_NODES * D)   // 12,900,000  (pair-packed weights)
#define OFF_SUM    (OFF_WPK + KTOT * D)       // 12,949,152
#define OFF_SUMSQ  (OFF_SUM + D)
#define OFF_SCALE  (OFF_SUMSQ + D)
#define OFF_SHIFT  (OFF_SCALE + D)
#define WS_FLOATS  (OFF_SHIFT + D)            // ~51.8 MB

// ------------------------------------------------------------------
__global__ void cheb_zero(float* ws, int n) {
    int i = blockIdx.x * blockDim.x + threadIdx.x;
    if (i < n) ws[i] = 0.0f;
}

// in-degree via float atomics
__global__ void cheb_deg(const int* __restrict__ edst, float* __restrict__ deg) {
    int e = blockIdx.x * blockDim.x + threadIdx.x;
    if (e < N_EDGES) atomicAdd(&deg[edst[e]], 1.0f);
}

// norm = clamp(deg,1)^-0.5
__global__ void cheb_norm(const float* __restrict__ deg, float* __restrict__ norm) {
    int i = blockIdx.x * blockDim.x + threadIdx.x;
    if (i < N_NODES) {
        float d = deg[i];
        d = d < 1.0f ? 1.0f : d;
        norm[i] = rsqrtf(d);
    }
}

// scatter-add: agg[dst] += src_mat[src] * scale(src)
// pass==0: scale = norm[src]           (msg = feature*norm)
// pass==1: scale = norm[src]^2         (msg = (agg1*norm)*norm, agg1 raw)
__global__ void cheb_scatter(const float* __restrict__ src_mat,
                             const float* __restrict__ norm,
                             const int* __restrict__ esrc,
                             const int* __restrict__ edst,
                             float* __restrict__ agg, int pass) {
    int t = blockIdx.x * blockDim.x + threadIdx.x;
    int e = t >> 5;
    if (e >= N_EDGES) return;
    int sub = (t & 31) * 4;
    int s = esrc[e];
    int d = edst[e];
    float sc = norm[s];
    if (pass) sc *= sc;
    float4 v = *(const float4*)(src_mat + s * D + sub);
    float* o = agg + d * D + sub;
    atomicAdd(o + 0, v.x * sc);
    atomicAdd(o + 1, v.y * sc);
    atomicAdd(o + 2, v.z * sc);
    atomicAdd(o + 3, v.w * sc);
}

// Folded weights, pair-packed for b64 B-fragment loads:
//   Wcat rows 0..127 = W1-W3, 128..255 = -W2, 256..383 = 2*W3
//   Wpk[p*2*D + n*2 + j] = Wcat[2p+j][n]   (p = k-row pair, n = out col, j = 0/1)
__global__ void cheb_wcat(const float* __restrict__ W1, const float* __restrict__ W2,
                          const float* __restrict__ W3, float* __restrict__ Wpk) {
    int i = blockIdx.x * blockDim.x + threadIdx.x;
    if (i >= KTOT * D) return;
    int row = i / D, col = i % D;
    float v;
    if (row < 128)       v = W1[row * D + col] - W3[row * D + col];
    else if (row < 256)  v = -W2[(row - 128) * D + col];
    else                 v = 2.0f * W3[(row - 256) * D + col];
    Wpk[(row >> 1) * (2 * D) + col * 2 + (row & 1)] = v;
}

// WMMA GEMM: h = [feature | agg1*norm | agg2*norm] @ Wcat + bias, then * snorm
// One wave per 16-node tile, 8 N-tiles of 16 cols => 16x128 output per wave.
// A fragment (f32 16x4): lane m=lane&15, kh=lane>>4; a = {A[m][2kh], A[m][2kh+1]}
// B fragment (f32 4x16): b = {W[k0][n], W[k0+1][n]} = one contiguous v2f in Wpk
__global__ void __launch_bounds__(128)
cheb_gemm(const float* __restrict__ feat, const float* __restrict__ agg1,
          const float* __restrict__ agg2, const float* __restrict__ norm,
          const float* __restrict__ Wpk, const float* __restrict__ bias,
          const float* __restrict__ snorm, float* __restrict__ out) {
    int wave = threadIdx.x >> 5;
    int tile = blockIdx.x * 4 + wave;
    if (tile >= NTILES) return;
    int lane = threadIdx.x & 31;
    int m  = lane & 15;
    int kh = lane >> 4;
    int row0 = tile * 16;
    float nrm = norm[row0 + m];

    v8f acc[8] = {};

    const float* Aseg[3] = { feat + row0 * D, agg1 + row0 * D, agg2 + row0 * D };
    float scl[3] = { 1.0f, nrm, nrm };

#pragma unroll
    for (int seg = 0; seg < 3; ++seg) {
        const float* A = Aseg[seg];
        float sc = scl[seg];
        for (int kk = 0; kk < D; kk += 4) {
            int k0 = kk + 2 * kh;            // lane's starting k-row
            v2f a = *(const v2f*)(A + m * D + k0);
            a.x *= sc;
            a.y *= sc;
            // pair index for this lane: seg*64 + kk/2 + kh
            const float* Wp = Wpk + (seg * (D / 2) + (kk >> 1) + kh) * (2 * D);
#pragma unroll
            for (int t = 0; t < 8; ++t) {
                v2f b = *(const v2f*)(Wp + (t * 16 + m) * 2);
                acc[t] = __builtin_amdgcn_wmma_f32_16x16x4_f32(
                    false, a, false, b, (short)0, acc[t], false, false);
            }
        }
    }

    // epilogue: + bias, * snorm ; C/D layout: VGPR r <-> M = r + 8*kh, N = lane&15
#pragma unroll
    for (int r = 0; r < 8; ++r) {
        int nodeR = row0 + r + 8 * kh;
        float sn = snorm[nodeR];
#pragma unroll
        for (int t = 0; t < 8; ++t) {
            int col = t * 16 + m;
            out[nodeR * D + col] = (acc[t][r] + bias[col]) * sn;
        }
    }
}

// per-feature sum / sumsq over nodes (128 threads = 128 features)
__global__ void __launch_bounds__(128)
cheb_bnred(const float* __restrict__ h, float* __restrict__ sum,
           float* __restrict__ sumsq) {
    int d = threadIdx.x;
    int n0 = blockIdx.x * 250;
    float s = 0.0f, s2 = 0.0f;
    for (int i = 0; i < 250; ++i) {
        float v = h[(n0 + i) * D + d];
        s += v;
        s2 += v * v;
    }
    atomicAdd(&sum[d], s);
    atomicAdd(&sumsq[d], s2);
}

// fold mean/var/gamma/beta into fused scale/shift
__global__ void cheb_stats(const float* __restrict__ sum, const float* __restrict__ sumsq,
                           const float* __restrict__ gamma, const float* __restrict__ beta,
                           float* __restrict__ scale, float* __restrict__ shift) {
    int d = threadIdx.x;
    if (d >= D) return;
    float mean = sum[d] * (1.0f / N_NODES);
    float var  = sumsq[d] * (1.0f / N_NODES) - mean * mean;
    float inv  = rsqrtf(var + BN_EPS);
    float sc   = inv * gamma[d];
    scale[d] = sc;
    shift[d] = beta[d] - mean * sc;
}

// out = feature + relu(h*scale + shift)   (h lives in d_out, rewritten in place)
__global__ void cheb_final(const float* __restrict__ feat, const float* __restrict__ scale,
                           const float* __restrict__ shift, float* __restrict__ out) {
    int t = blockIdx.x * blockDim.x + threadIdx.x;
    if (t >= N_NODES * 32) return;
    int node = t >> 5;
    int c = (t & 31) * 4;
    float4 hv = *(const float4*)(out + node * D + c);
    float4 fv = *(const float4*)(feat + node * D + c);
    float4 sc = *(const float4*)(scale + c);
    float4 sh = *(const float4*)(shift + c);
    float4 y;
    y.x = fmaf(hv.x, sc.x, sh.x); y.x = y.x > 0.0f ? y.x : 0.0f;
    y.y = fmaf(hv.y, sc.y, sh.y); y.y = y.y > 0.0f ? y.y : 0.0f;
    y.z = fmaf(hv.z, sc.z, sh.z); y.z = y.z > 0.0f ? y.z : 0.0f;
    y.w = fmaf(hv.w, sc.w, sh.w); y.w = y.w > 0.0f ? y.w : 0.0f;
    y.x += fv.x; y.y += fv.y; y.z += fv.z; y.w += fv.w;
    *(float4*)(out + node * D + c) = y;
}

// ------------------------------------------------------------------
extern "C" void kernel_launch(void* const* d_in, const int* in_sizes, int n_in,
                              void* d_out, int out_size, void* d_ws, size_t ws_size,
                              hipStream_t stream) {
    const float* feature = (const float*)d_in[0];
    const float* snorm   = (const float*)d_in[1];
    const float* W1      = (const float*)d_in[2];
    const float* W2      = (const float*)d_in[3];
    const float* W3      = (const float*)d_in[4];
    const float* bias    = (const float*)d_in[5];
    const float* gamma   = (const float*)d_in[6];
    const float* beta    = (const float*)d_in[7];
    const int*   esrc    = (const int*)d_in[8];
    const int*   edst    = (const int*)d_in[9];
    float* out = (float*)d_out;
    float* ws  = (float*)d_ws;

    float* deg   = ws + OFF_DEG;
    float* norm  = ws + OFF_NORM;
    float* agg1  = ws + OFF_AGG1;
    float* agg2  = ws + OFF_AGG2;
    float* Wpk   = ws + OFF_WPK;
    float* sum   = ws + OFF_SUM;
    float* sumsq = ws + OFF_SUMSQ;
    float* scale = ws + OFF_SCALE;
    float* shift = ws + OFF_SHIFT;

    // zero deg/agg1/agg2/sum/sumsq (zero the whole used region)
    cheb_zero<<<(WS_FLOATS + 255) / 256, 256, 0, stream>>>(ws, WS_FLOATS);

    // degree + norm
    cheb_deg<<<(N_EDGES + 255) / 256, 256, 0, stream>>>(edst, deg);
    cheb_norm<<<(N_NODES + 255) / 256, 256, 0, stream>>>(deg, norm);

    // message-pass 1: agg1[dst] += feature[src]*norm[src]
    int sgrid = (N_EDGES * 32 + 255) / 256;
    cheb_scatter<<<sgrid, 256, 0, stream>>>(feature, norm, esrc, edst, agg1, 0);
    // message-pass 2: agg2[dst] += (agg1[src]*norm[src])*norm[src]
    cheb_scatter<<<sgrid, 256, 0, stream>>>(agg1, norm, esrc, edst, agg2, 1);

    // fold weights into pair-packed Wpk
    cheb_wcat<<<(KTOT * D + 255) / 256, 256, 0, stream>>>(W1, W2, W3, Wpk);

    // WMMA GEMM + bias + snorm -> h (stored in d_out)
    cheb_gemm<<<(NTILES + 3) / 4, 128, 0, stream>>>(feature, agg1, agg2, norm,
                                                    Wpk, bias, snorm, out);

    // batchnorm statistics + fused scale/shift
    cheb_bnred<<<N_NODES / 250, 128, 0, stream>>>(out, sum, sumsq);
    cheb_stats<<<1, 128, 0, stream>>>(sum, sumsq, gamma, beta, scale, shift);

    // residual + relu, in place on d_out
    cheb_final<<<(N_NODES * 32 + 255) / 256, 256, 0, stream>>>(feature, scale, shift, out);
}